// HierarchicalLLLoss_2671469658317
// MI455X (gfx1250) — compile-verified
//
#include <hip/hip_runtime.h>

// Problem constants (match reference)
#define NB     1024      // batch
#define CL     2048      // leaf classes
#define EE     12        // edges per path
#define WAVES  8         // waves per block (256 threads, wave32)
#define CSTRIDE (CL / WAVES)     // 256 c-values per wave
#define UNROLL  4                // 4 K-steps (16 c) per macro-iteration
#define NITER  (CSTRIDE / (4 * UNROLL))   // 16 macro-iterations

typedef float v2f __attribute__((ext_vector_type(2)));
typedef float v8f __attribute__((ext_vector_type(8)));

// One block per sample. Wave w reduces c in [w*256, w*256+256) with
// V_WMMA_F32_16X16X4_F32:  A[m=e, k=c] = onehot[t][c][e],  B[k,n] = x[c]
// (broadcast across N, so the N-lane mapping is irrelevant).  D accumulates
// num[e]/den[e] across the whole c loop in 8 VGPRs.
//
// Rows e=12..15 of A are loaded from a clamped (duplicate e=0) address: a
// matmul never mixes A-rows, so they pollute only D rows 12..15, which are
// never read.  No masking multiply needed -> loads feed WMMA directly.
__global__ __launch_bounds__(256) void hll_per_sample_kernel(
    const float* __restrict__ x,          // [NB, CL]
    const int*   __restrict__ target,     // [NB]
    const float* __restrict__ onum,       // [C_ALL, CL, EE]
    const float* __restrict__ oden,       // [C_ALL, CL, EE]
    const float* __restrict__ wts,        // [C_ALL, EE]
    float*       __restrict__ per_sample) // [NB]
{
    __shared__ float sN[WAVES][16];
    __shared__ float sD[WAVES][16];
    __shared__ float sV[16];

    const int b    = blockIdx.x;
    const int tid  = threadIdx.x;
    const int wv   = tid >> 5;
    const int lane = tid & 31;
    const int t    = target[b];

    const float* __restrict__ nrow = onum + (size_t)t * (CL * EE);
    const float* __restrict__ drow = oden + (size_t)t * (CL * EE);
    const float* __restrict__ xrow = x    + (size_t)b * CL;

    // A-matrix 16x4 f32 lane layout: lanes 0-15 hold K=0 (vgpr0), K=1 (vgpr1);
    // lanes 16-31 hold K=2,3.  m (= e) is lane&15, K base is (lane>>4)*2.
    const int e  = lane & 15;
    const int g  = lane >> 4;
    const int ec = (e < EE) ? e : 0;   // in-bounds addr for junk rows, uniform flow

    v8f accN = {0.f,0.f,0.f,0.f,0.f,0.f,0.f,0.f};
    v8f accD = {0.f,0.f,0.f,0.f,0.f,0.f,0.f,0.f};

    // Per-wave base pointers; all in-loop offsets are compile-time constants so
    // loads fold into IOFFSET immediates off a single advancing base.
    const int cBeg = wv * CSTRIDE;
    const float* pn = nrow + (size_t)(cBeg + 2 * g) * EE + ec;
    const float* pd = drow + (size_t)(cBeg + 2 * g) * EE + ec;
    const float* px = xrow + cBeg + 2 * g;

    for (int it = 0; it < NITER; ++it) {
        // Uniform prefetch of the next 16-c region (768 B/wave/array): three
        // 256 B-spaced bases x lane spread cover all 6 cachelines.
        const int pf = (it + 1 < NITER) ? (4 * UNROLL * EE) : 0;
        __builtin_prefetch(pn + pf,       0, 1);
        __builtin_prefetch(pn + pf +  64, 0, 1);
        __builtin_prefetch(pn + pf + 128, 0, 1);
        __builtin_prefetch(pd + pf,       0, 1);
        __builtin_prefetch(pd + pf +  64, 0, 1);
        __builtin_prefetch(pd + pf + 128, 0, 1);

        #pragma unroll
        for (int u = 0; u < UNROLL; ++u) {
            v2f An = { pn[(u * 4    ) * EE], pn[(u * 4 + 1) * EE] };
            v2f Ad = { pd[(u * 4    ) * EE], pd[(u * 4 + 1) * EE] };
            v2f Bx = { px[u * 4],            px[u * 4 + 1]        };

            // 8 args: (neg_a, A, neg_b, B, c_mod, C, reuse_a, reuse_b)
            accN = __builtin_amdgcn_wmma_f32_16x16x4_f32(false, An, false, Bx,
                                                         (short)0, accN, false, false);
            accD = __builtin_amdgcn_wmma_f32_16x16x4_f32(false, Ad, false, Bx,
                                                         (short)0, accD, false, false);
        }

        pn += 4 * UNROLL * EE;
        pd += 4 * UNROLL * EE;
        px += 4 * UNROLL;
    }

    // C/D layout: lanes 0-15 col n=lane, vgpr v -> M=v; lanes 16-31 -> M=v+8.
    // All columns identical, so lane 0 yields e=0..7 and lane 16 yields e=8..15
    // (e>=12 is junk and is never read below).
    if (lane == 0) {
        #pragma unroll
        for (int v = 0; v < 8; ++v) { sN[wv][v] = accN[v]; sD[wv][v] = accD[v]; }
    } else if (lane == 16) {
        #pragma unroll
        for (int v = 0; v < 8; ++v) { sN[wv][8 + v] = accN[v]; sD[wv][8 + v] = accD[v]; }
    }
    __syncthreads();

    if (tid < EE) {
        float n = 0.f, d = 0.f;
        #pragma unroll
        for (int w = 0; w < WAVES; ++w) { n += sN[w][tid]; d += sD[w][tid]; }
        const float val = (n != 0.f) ? -__logf(n / d) : 0.f;
        sV[tid] = wts[(size_t)t * EE + tid] * val;
    }
    __syncthreads();

    if (tid == 0) {
        float s = 0.f;
        #pragma unroll
        for (int i = 0; i < EE; ++i) s += sV[i];
        per_sample[b] = s;
    }
}

// Deterministic single-block mean reduction (no float atomics).
__global__ __launch_bounds__(256) void hll_reduce_kernel(
    const float* __restrict__ per_sample, float* __restrict__ out, int nb)
{
    __shared__ float ss[256];
    const int tid = threadIdx.x;
    float s = 0.f;
    for (int i = tid; i < nb; i += 256) s += per_sample[i];
    ss[tid] = s;
    __syncthreads();
    for (int off = 128; off > 0; off >>= 1) {
        if (tid < off) ss[tid] += ss[tid + off];
        __syncthreads();
    }
    if (tid == 0) out[0] = ss[0] / (float)nb;
}

extern "C" void kernel_launch(void* const* d_in, const int* in_sizes, int n_in,
                              void* d_out, int out_size, void* d_ws, size_t ws_size,
                              hipStream_t stream) {
    // setup_inputs() order: inputs, target, onehot_num, onehot_den, weights
    const float* x      = (const float*)d_in[0];
    const int*   target = (const int*)  d_in[1];
    const float* onum   = (const float*)d_in[2];
    const float* oden   = (const float*)d_in[3];
    const float* wts    = (const float*)d_in[4];

    const int nb = in_sizes[1];              // batch size from target length
    float* per_sample = (float*)d_ws;        // nb floats of scratch

    hll_per_sample_kernel<<<nb, 256, 0, stream>>>(x, target, onum, oden, wts, per_sample);
    hll_reduce_kernel<<<1, 256, 0, stream>>>(per_sample, (float*)d_out, nb);
}